// FFTBlock_48412871360934
// MI455X (gfx1250) — compile-verified
//
#include <hip/hip_runtime.h>
#include <hip/hip_bf16.h>
#include <math.h>

// ---------------------------------------------------------------------------
// FNO block on MI455X (gfx1250), wave32, V_WMMA_F32_16X16X4_F32 throughout.
// Pipeline (all GEMMs via f32 WMMA; large contiguous LDS staging via TDM):
//   k_twiddle : build DFT / synthesis matrices in ws (reads fft_scale)
//   k_stats   : per-channel sum/sumsq (batch-norm batch statistics)
//   k_bnp     : fold stats+gamma+beta into per-channel scale/shift
//   k_fft1    : partial forward W-DFT   x[b,c,h,:256] -> Y1[b,c,(re|im),h,kx<32]
//   k_fft2    : forward H-DFT           Y1 -> X2[b,c,(re|im)*32+ky, kx]   [TDM]
//   k_mix     : per-mode complex channel mix (M=16 packs 8 batches x re/im)
//   k_inv1    : inverse H-DFT           XM -> Y3[b,o,(re|im),h,kx]        [TDM]
//   k_final   : inverse-W synthesis GEMM + fused BN+GELU+1x1-conv GEMM
//               + identity skip + bias  -> out
// ---------------------------------------------------------------------------

#define B_ 8
#define C_ 64
#define O_ 64
#define H_ 256
#define W_ 256
#define M_ 32

typedef __attribute__((ext_vector_type(2))) float v2f;
typedef __attribute__((ext_vector_type(8))) float v8f;
typedef __attribute__((ext_vector_type(4))) unsigned int u32x4;
typedef __attribute__((ext_vector_type(8))) int i32x8;
typedef __attribute__((ext_vector_type(4))) int i32x4;

__device__ __forceinline__ v8f vzero8() {
  v8f z = {0.f, 0.f, 0.f, 0.f, 0.f, 0.f, 0.f, 0.f};
  return z;
}

// ---------------------------------------------------------------------------
// Tensor Data Mover: 1-D contiguous copy global -> LDS (nelem f32, <= 65535).
// D# packing per CDNA5 ISA:
//   g0[0]=count=1; g0[1]=lds byte addr; g0[2..3]=global addr[56:0], type=2
//   g1[0]=data_size 4B; tensor_dim0=tile_dim0=nelem; tensor_dim1=tile_dim1=1
// This toolchain exposes the 6-arg builtin (clang-23 / therock-10.0 form):
//   (u32x4 g0, i32x8 g1, i32x4 g2, i32x4 g3, i32x8 g4, i32 cpol)
// Groups 2-4 are unused for a <=2-D tensor -> zero-filled.
// Issue from ONE wave, s_wait_tensorcnt 0, then barrier for the workgroup.
// ---------------------------------------------------------------------------
__device__ __forceinline__ void tdm_load_1d(const float* gsrc, void* lds_dst,
                                            unsigned nelem) {
  unsigned long long ga = (unsigned long long)(uintptr_t)gsrc;
  unsigned ldsoff = (unsigned)(uintptr_t)lds_dst;  // low 32 bits = LDS offset
  u32x4 g0;
  g0[0] = 1u;                                        // count=1 (valid user D#)
  g0[1] = ldsoff;
  g0[2] = (unsigned)(ga & 0xFFFFFFFFu);
  g0[3] = (unsigned)((ga >> 32) & 0x01FFFFFFu) | 0x80000000u;  // type=2
  i32x8 g1;
  g1[0] = 0x00020000;                   // workgroup_mask=0, data_size=2 (4B)
  g1[1] = (int)((nelem & 0xFFFFu) << 16);          // tensor_dim0[15:0]
  g1[2] = (int)((1u << 16) | (nelem >> 16));       // tensor_dim1=1 | dim0 hi
  g1[3] = (int)(nelem << 16);                      // tile_dim0 = nelem
  g1[4] = 1;                                       // tile_dim1=1, tile_dim2=0
  g1[5] = (int)nelem;                              // tensor_dim0_stride
  g1[6] = 0;
  g1[7] = 0;
  i32x4 gz = {0, 0, 0, 0};
  i32x8 gz8 = {0, 0, 0, 0, 0, 0, 0, 0};
  __builtin_amdgcn_tensor_load_to_lds(g0, g1, gz, gz, gz8, 0);
}

// 16x16 output tile, K-loop in steps of 4 using V_WMMA_F32_16X16X4_F32.
// A: row-major [16 x K] with row stride lda (LDS or global)
// B: row-major [K x >=16] with row stride ldb, col window starts at Bm
// Per-ISA lane layout: A lane(r=lane&15) holds K = 2*(lane>>4)+{0,1};
//                      B lane(c=lane&15) holds K rows 2*(lane>>4)+{0,1};
//                      C/D element v is row 8*(lane>>4)+v, col lane&15.
__device__ __forceinline__ v8f wmma_tile(const float* __restrict__ A, int lda,
                                         const float* __restrict__ Bm, int ldb,
                                         int K, v8f acc) {
  const int lane = threadIdx.x & 31;
  const int r = lane & 15;
  const int koff = (lane >> 4) * 2;
  for (int k = 0; k < K; k += 4) {
    v2f a, b;
    a.x = A[r * lda + k + koff];
    a.y = A[r * lda + k + koff + 1];
    b.x = Bm[(k + koff) * ldb + r];
    b.y = Bm[(k + koff + 1) * ldb + r];
    acc = __builtin_amdgcn_wmma_f32_16x16x4_f32(false, a, false, b,
                                                (short)0, acc, false, false);
  }
  return acc;
}

// ---------------------------------------------------------------------------
// Twiddle / synthesis matrix construction.
//   FW [256 w][64 n]   n=q*32+kx : q0=cos, q1=-sin     (e^{-i t}, forward W)
//   T2 [64 m][512 k]   m=p*32+ky, k=q*256+h            (forward H, complex)
//   T3 [512 m][64 k]   m=p*256+h, k=q*32+ky            (inverse H, e^{+i t})
//   G  [64 n][256 w]   n=q*32+kx : synthesis of irfft-W, scale & 2x folded in
// ---------------------------------------------------------------------------
__global__ void k_twiddle(float* __restrict__ FW, float* __restrict__ T2,
                          float* __restrict__ T3, float* __restrict__ G,
                          const float* __restrict__ fscale) {
  const float STEP = 6.283185307179586f / 256.f;
  const int stride = gridDim.x * blockDim.x;
  const int tid = blockIdx.x * blockDim.x + threadIdx.x;

  for (int i = tid; i < 256 * 64; i += stride) {
    int w = i >> 6, n = i & 63, q = n >> 5, kx = n & 31;
    float ang = STEP * (float)((kx * w) & 255);
    FW[i] = q ? -sinf(ang) : cosf(ang);
  }
  for (int i = tid; i < 64 * 512; i += stride) {
    int m = i >> 9, k = i & 511;
    int p = m >> 5, ky = m & 31, q = k >> 8, h = k & 255;
    float ang = STEP * (float)((ky * h) & 255);
    float c = cosf(ang), s = sinf(ang);
    // e^{-i t}(yr + i yi): re = c*yr + s*yi ; im = c*yi - s*yr
    T2[i] = (p == 0) ? (q == 0 ? c : s) : (q == 0 ? -s : c);
  }
  for (int i = tid; i < 512 * 64; i += stride) {
    int m = i >> 6, k = i & 63;
    int p = m >> 8, h = m & 255, q = k >> 5, ky = k & 31;
    float ang = STEP * (float)((ky * h) & 255);
    float c = cosf(ang), s = sinf(ang);
    // e^{+i t}(xr + i xi): re = c*xr - s*xi ; im = s*xr + c*xi
    T3[i] = (p == 0) ? (q == 0 ? c : -s) : (q == 0 ? s : c);
  }
  const float s0 = fscale[0] / (float)(H_ * W_);
  for (int i = tid; i < 64 * 256; i += stride) {
    int n = i >> 8, w = i & 255, q = n >> 5, kx = n & 31;
    float amp = (kx == 0) ? s0 : 2.f * s0;  // irfft: factor 2 except DC
    float ang = STEP * (float)((kx * w) & 255);
    float val = q ? -amp * sinf(ang) : amp * cosf(ang);
    if (q == 1 && kx == 0) val = 0.f;  // c2r ignores Im of DC bin
    G[i] = val;
  }
}

// ---------------------------------------------------------------------------
__global__ void k_stats(const float* __restrict__ x, float* __restrict__ stats) {
  const int c = blockIdx.x, b = blockIdx.y;
  const float* p = x + ((size_t)(b * C_ + c)) * H_ * W_;
  float s = 0.f, s2 = 0.f;
  for (int i = threadIdx.x; i < H_ * W_; i += blockDim.x) {
    float v = p[i];
    s += v;
    s2 += v * v;
  }
  __shared__ float sh0[256];
  __shared__ float sh1[256];
  sh0[threadIdx.x] = s;
  sh1[threadIdx.x] = s2;
  __syncthreads();
  for (int o = 128; o > 0; o >>= 1) {
    if (threadIdx.x < (unsigned)o) {
      sh0[threadIdx.x] += sh0[threadIdx.x + o];
      sh1[threadIdx.x] += sh1[threadIdx.x + o];
    }
    __syncthreads();
  }
  if (threadIdx.x == 0) {
    atomicAdd(&stats[c], sh0[0]);
    atomicAdd(&stats[64 + c], sh1[0]);
  }
}

__global__ void k_bnp(const float* __restrict__ stats,
                      const float* __restrict__ gamma,
                      const float* __restrict__ beta, float* __restrict__ bnp) {
  int c = threadIdx.x;
  const float n = (float)(B_ * H_ * W_);
  float mean = stats[c] / n;
  float var = stats[64 + c] / n - mean * mean;
  float sc = gamma[c] * rsqrtf(var + 1e-5f);
  bnp[c] = sc;
  bnp[64 + c] = beta[c] - mean * sc;
}

// ---------------------------------------------------------------------------
// Stage 1: forward W-DFT.  Per (b,c), 16-row h-tile:
//   out[16 h][64 n] = x[16 h][256 w] @ FW[256 w][64 n]
// ---------------------------------------------------------------------------
__global__ void k_fft1(const float* __restrict__ x, const float* __restrict__ FW,
                       float* __restrict__ Y1) {
  __shared__ float Ash[16][W_ + 1];  // padded: avoid A-side bank conflicts
  __shared__ float Bsh[W_][64];
  const int bc = blockIdx.x, h0 = blockIdx.y * 16;
  const float* xp = x + ((size_t)bc) * H_ * W_ + (size_t)h0 * W_;
  for (int i = threadIdx.x; i < 16 * W_; i += blockDim.x) {
    Ash[i >> 8][i & 255] = xp[i];
    __builtin_prefetch(xp + i + 16 * W_, 0, 0);  // global_prefetch_b8 next tile
  }
  // FW is 64KB and contiguous: stage it with the Tensor Data Mover.
  if (threadIdx.x < 32) {
    tdm_load_1d(FW, &Bsh[0][0], W_ * 64);
    __builtin_amdgcn_s_wait_tensorcnt((short)0);
  }
  __syncthreads();

  const int wave = threadIdx.x >> 5, lane = threadIdx.x & 31;
  v8f acc = vzero8();
  acc = wmma_tile(&Ash[0][0], W_ + 1, &Bsh[0][wave * 16], 64, W_, acc);

  const int n = wave * 16 + (lane & 15);
  const int q = n >> 5, kx = n & 31;
  float* yb = Y1 + ((size_t)(bc * 2 + q)) * H_ * 32 + kx;
  const int rbase = (lane >> 4) * 8;
#pragma unroll
  for (int v = 0; v < 8; v++) yb[(size_t)(h0 + rbase + v) * 32] = acc[v];
}

// ---------------------------------------------------------------------------
// Stage 2: forward H-DFT. Per (b,c): X2[64 m][32 kx] = T2[64][512] @ Y1bc[512][32]
// Y1bc is a contiguous 64KB block -> TDM async copy into LDS.
// ---------------------------------------------------------------------------
__global__ void k_fft2(const float* __restrict__ Y1, const float* __restrict__ T2,
                       float* __restrict__ X2) {
  __shared__ float Bsh[512][32];
  const int bc = blockIdx.x;
  const float* yp = Y1 + (size_t)bc * 512 * 32;
  if (threadIdx.x < 32) {
    tdm_load_1d(yp, &Bsh[0][0], 512 * 32);
    __builtin_amdgcn_s_wait_tensorcnt((short)0);
  }
  __syncthreads();

  const int wave = threadIdx.x >> 5, lane = threadIdx.x & 31;
  float* xo = X2 + (size_t)bc * 64 * 32;
  for (int nt = 0; nt < 2; nt++) {
    v8f acc = vzero8();
    acc = wmma_tile(T2 + (size_t)(wave * 16) * 512, 512, &Bsh[0][nt * 16], 32, 512, acc);
    const int col = nt * 16 + (lane & 15);
    const int rb = wave * 16 + (lane >> 4) * 8;
#pragma unroll
    for (int v = 0; v < 8; v++) xo[(size_t)(rb + v) * 32 + col] = acc[v];
  }
}

// ---------------------------------------------------------------------------
// Stage 3: per-mode complex channel mixing.
// M=16 rows pack (b, re/im); K=128 packs (re/im, c):
//   out_re[b,o] = sum_c Xr*Wr - Xi*Wi ;  out_im = sum_c Xi*Wr + Xr*Wi
// ---------------------------------------------------------------------------
__global__ void k_mix(const float* __restrict__ X2, const float* __restrict__ Wr,
                      const float* __restrict__ Wi, float* __restrict__ XM) {
  __shared__ float Ash[16][129];  // padded
  __shared__ float Bsh[128][64];
  const int kx = blockIdx.x, ky = blockIdx.y;

  for (int i = threadIdx.x; i < 16 * 128; i += blockDim.x) {
    int m = i >> 7, k = i & 127;
    int b = m >> 1, p = m & 1, q = k >> 6, c = k & 63;
    float xr = X2[((size_t)(b * 64 + c) * 64 + ky) * 32 + kx];
    float xi = X2[((size_t)(b * 64 + c) * 64 + 32 + ky) * 32 + kx];
    float v = (p == 0) ? ((q == 0) ? xr : -xi) : ((q == 0) ? xi : xr);
    Ash[m][k] = v;
  }
  for (int i = threadIdx.x; i < 128 * 64; i += blockDim.x) {
    int k = i >> 6, o = i & 63;
    int q = k >> 6, c = k & 63;
    const float* Wp = q ? Wi : Wr;
    Bsh[k][o] = Wp[((size_t)(c * 64 + o) * 32 + ky) * 32 + kx];
  }
  __syncthreads();

  const int wave = threadIdx.x >> 5, lane = threadIdx.x & 31;
  v8f acc = vzero8();
  acc = wmma_tile(&Ash[0][0], 129, &Bsh[0][wave * 16], 64, 128, acc);

  const int o = wave * 16 + (lane & 15);
#pragma unroll
  for (int v = 0; v < 8; v++) {
    int m = (lane >> 4) * 8 + v;
    int b = m >> 1, p = m & 1;
    XM[(((size_t)(b * 64 + o) * 2 + p) * 32 + ky) * 32 + kx] = acc[v];
  }
}

// ---------------------------------------------------------------------------
// Stage 4: inverse H-DFT. Per (b,o): Y3[512 m][32 kx] = T3[512][64] @ XMbo[64][32]
// XMbo is a contiguous 8KB block -> TDM async copy into LDS.
// ---------------------------------------------------------------------------
__global__ void k_inv1(const float* __restrict__ XM, const float* __restrict__ T3,
                       float* __restrict__ Y3) {
  __shared__ float Bsh[64][32];
  const int bo = blockIdx.x;
  const float* xp = XM + (size_t)bo * 64 * 32;
  if (threadIdx.x < 32) {
    tdm_load_1d(xp, &Bsh[0][0], 64 * 32);
    __builtin_amdgcn_s_wait_tensorcnt((short)0);
  }
  __syncthreads();

  const int wave = threadIdx.x >> 5, lane = threadIdx.x & 31;
  float* yo = Y3 + (size_t)bo * 512 * 32;
  for (int mt = wave; mt < 32; mt += 4) {
    for (int nt = 0; nt < 2; nt++) {
      v8f acc = vzero8();
      acc = wmma_tile(T3 + (size_t)(mt * 16) * 64, 64, &Bsh[0][nt * 16], 32, 64, acc);
      const int col = nt * 16 + (lane & 15);
      const int rb = mt * 16 + (lane >> 4) * 8;
#pragma unroll
      for (int v = 0; v < 8; v++) yo[(size_t)(rb + v) * 32 + col] = acc[v];
    }
  }
}

// ---------------------------------------------------------------------------
// Stage 5 (fused epilogue). Per (b,h):
//   out[64 o][256 w] = A1[64 o][64 n] @ G[64 n][256 w]          (inverse-W)
//                    + PW[64 o][64 i] @ gelu(bn(x))[64 i][256 w] (1x1 conv)
//                    + x[b,o,h,w] + proj_b[o]                    (skip + bias)
// ---------------------------------------------------------------------------
__global__ void k_final(const float* __restrict__ x, const float* __restrict__ Y3,
                        const float* __restrict__ G, const float* __restrict__ PW,
                        const float* __restrict__ pb, const float* __restrict__ bnp,
                        float* __restrict__ out) {
  __shared__ float A1[64][65];   // padded
  __shared__ float Gg[64][W_];
  const int h = blockIdx.x, b = blockIdx.y;

  for (int i = threadIdx.x; i < 64 * 64; i += blockDim.x) {
    int o = i >> 6, n = i & 63;
    A1[o][n] = Y3[(((size_t)(b * 64 + o) * 2 + (n >> 5)) * 256 + h) * 32 + (n & 31)];
  }
  for (int i = threadIdx.x; i < 64 * W_; i += blockDim.x) {
    int c = i >> 8, w = i & 255;
    float xv = x[(((size_t)(b * 64 + c)) * 256 + h) * 256 + w];
    float xn = xv * bnp[c] + bnp[64 + c];
    Gg[c][w] = 0.5f * xn * (1.f + erff(xn * 0.70710678118654752f));  // exact GELU
  }
  __syncthreads();

  const int wave = threadIdx.x >> 5, lane = threadIdx.x & 31;
  for (int ot = 0; ot < 4; ot++) {
    for (int wt = wave; wt < 16; wt += 4) {
      v8f acc = vzero8();
      acc = wmma_tile(&A1[ot * 16][0], 65, G + wt * 16, W_, 64, acc);           // spectral
      acc = wmma_tile(PW + (size_t)(ot * 16) * 64, 64, &Gg[0][wt * 16], W_, 64, acc);  // conv
#pragma unroll
      for (int v = 0; v < 8; v++) {
        int o = ot * 16 + (lane >> 4) * 8 + v;
        int w = wt * 16 + (lane & 15);
        size_t idx = (((size_t)(b * 64 + o)) * 256 + h) * 256 + w;
        out[idx] = acc[v] + x[idx] + pb[o];
      }
    }
  }
}

// ---------------------------------------------------------------------------
extern "C" void kernel_launch(void* const* d_in, const int* in_sizes, int n_in,
                              void* d_out, int out_size, void* d_ws, size_t ws_size,
                              hipStream_t stream) {
  (void)in_sizes; (void)n_in; (void)out_size; (void)ws_size;
  const float* x      = (const float*)d_in[0];
  const float* wr     = (const float*)d_in[1];
  const float* wi     = (const float*)d_in[2];
  const float* fscale = (const float*)d_in[3];
  const float* gamma  = (const float*)d_in[4];
  const float* beta   = (const float*)d_in[5];
  const float* pw     = (const float*)d_in[6];
  const float* pb     = (const float*)d_in[7];
  float* out = (float*)d_out;

  float* ws    = (float*)d_ws;
  float* stats = ws;                 // 128
  float* bnp   = ws + 128;           // 128
  float* FW    = ws + 256;           // 256*64
  float* T2    = FW + 256 * 64;      // 64*512
  float* T3    = T2 + 64 * 512;      // 512*64
  float* G     = T3 + 512 * 64;      // 64*256
  float* Y1    = G + 64 * 256;       // 8*64*2*256*32  (reused as Y3)
  float* X2    = Y1 + (size_t)B_ * C_ * 2 * H_ * 32;  // 8*64*64*32
  float* XM    = X2 + (size_t)B_ * C_ * 64 * 32;      // 8*64*2*32*32

  (void)hipMemsetAsync(stats, 0, 128 * sizeof(float), stream);
  k_twiddle<<<64, 256, 0, stream>>>(FW, T2, T3, G, fscale);
  k_stats<<<dim3(C_, B_), 256, 0, stream>>>(x, stats);
  k_bnp<<<1, 64, 0, stream>>>(stats, gamma, beta, bnp);
  k_fft1<<<dim3(B_ * C_, H_ / 16), 128, 0, stream>>>(x, FW, Y1);
  k_fft2<<<B_ * C_, 128, 0, stream>>>(Y1, T2, X2);
  k_mix<<<dim3(M_, M_), 128, 0, stream>>>(X2, wr, wi, XM);
  k_inv1<<<B_ * O_, 128, 0, stream>>>(XM, T3, Y1 /* Y3 reuses Y1 */);
  k_final<<<dim3(H_, B_), 128, 0, stream>>>(x, Y1, G, pw, pb, bnp, out);
}